// GINConv_19645180412752
// MI455X (gfx1250) — compile-verified
//
#include <hip/hip_runtime.h>
#include <hip/hip_bf16.h>

typedef float v2f __attribute__((ext_vector_type(2)));
typedef float v8f __attribute__((ext_vector_type(8)));

#define D_FEAT 64
#define D_HID  128
#define WAVES_PER_BLOCK 4

// ---------------------------------------------------------------------------
// Kernel 1: agg[i] = (1 + eps) * x[i]
// ---------------------------------------------------------------------------
__global__ void gin_init_agg(const float* __restrict__ x,
                             const float* __restrict__ eps,
                             float* __restrict__ agg, int n) {
    int i = blockIdx.x * blockDim.x + threadIdx.x;
    if (i < n) agg[i] = (1.0f + eps[0]) * x[i];
}

// ---------------------------------------------------------------------------
// Kernel 2: per-edge scatter-add. 64 consecutive threads share one edge so the
// row/col loads broadcast; each thread does one fp32 global atomic (L2-resident).
// ---------------------------------------------------------------------------
__global__ void gin_scatter(const float* __restrict__ x,
                            const int* __restrict__ row,
                            const int* __restrict__ col,
                            float* __restrict__ agg,
                            long long n_items) {
    long long gid = (long long)blockIdx.x * blockDim.x + threadIdx.x;
    if (gid >= n_items) return;
    long long e = gid >> 6;          // edge id
    int f = (int)(gid & 63);         // feature id
    int r = row[e];
    int c = col[e];
    if (r != c) {                    // remove self loops
        unsafeAtomicAdd(&agg[(long long)r * D_FEAT + f],
                        x[(long long)c * D_FEAT + f]);
    }
}

// ---------------------------------------------------------------------------
// Kernel 3: fused MLP  y = relu(agg@W1 + b1)@W2 + b2  using V_WMMA_F32_16X16X4_F32.
// One wave32 per 16-node tile. X tile and hidden tile staged in padded LDS.
// ---------------------------------------------------------------------------
__global__ void __launch_bounds__(WAVES_PER_BLOCK * 32)
gin_mlp(const float* __restrict__ agg,
        const float* __restrict__ w1, const float* __restrict__ b1,
        const float* __restrict__ w2, const float* __restrict__ b2,
        float* __restrict__ out, int n_nodes) {
    // padded LDS tiles (pad +1 dword breaks the stride-64 bank conflict)
    __shared__ float lds_x[WAVES_PER_BLOCK][16 * (D_FEAT + 1)];   // 16x64 X tile
    __shared__ float lds_h[WAVES_PER_BLOCK][16 * (D_HID + 1)];    // 16x128 hidden tile

    const int lane = threadIdx.x & 31;
    const int wid  = threadIdx.x >> 5;
    const int hi   = lane >> 4;      // 0: lanes 0-15, 1: lanes 16-31
    const int l15  = lane & 15;

    const int node0 = (blockIdx.x * WAVES_PER_BLOCK + wid) * 16;
    if (node0 >= n_nodes) return;    // wave-uniform: EXEC stays all-ones below

    float* xt = &lds_x[wid][0];
    float* ht = &lds_h[wid][0];

    // load 16x64 X tile into LDS (coalesced, 32 floats per lane)
    for (int i = lane; i < 16 * D_FEAT; i += 32) {
        int r = i >> 6;              // node row 0..15
        int c = i & 63;              // feature 0..63
        xt[r * (D_FEAT + 1) + c] = agg[(long long)(node0 + r) * D_FEAT + c];
    }

    // ---- layer 1: h = relu(X @ W1 + b1), 8 N-tiles of 16, K = 64 ----
    for (int nt = 0; nt < D_HID / 16; ++nt) {
        v8f acc = {};
        const int n0 = nt * 16;
        for (int k0 = 0; k0 < D_FEAT; k0 += 4) {
            const int kb = k0 + 2 * hi;
            v2f a, b;
            a.x = xt[l15 * (D_FEAT + 1) + kb];
            a.y = xt[l15 * (D_FEAT + 1) + kb + 1];
            b.x = w1[(kb    ) * D_HID + n0 + l15];
            b.y = w1[(kb + 1) * D_HID + n0 + l15];
            acc = __builtin_amdgcn_wmma_f32_16x16x4_f32(
                false, a, false, b, (short)0, acc, false, false);
        }
        const float bias = b1[n0 + l15];
        for (int r = 0; r < 8; ++r) {
            float v = acc[r] + bias;
            v = v > 0.0f ? v : 0.0f;                         // ReLU
            ht[(r + 8 * hi) * (D_HID + 1) + n0 + l15] = v;   // stage hidden tile
        }
    }

    // ---- layer 2: y = h @ W2 + b2, 4 N-tiles of 16, K = 128 ----
    for (int nt = 0; nt < D_FEAT / 16; ++nt) {
        v8f acc = {};
        const int n0 = nt * 16;
        for (int k0 = 0; k0 < D_HID; k0 += 4) {
            const int kb = k0 + 2 * hi;
            v2f a, b;
            a.x = ht[l15 * (D_HID + 1) + kb];
            a.y = ht[l15 * (D_HID + 1) + kb + 1];
            b.x = w2[(kb    ) * D_FEAT + n0 + l15];
            b.y = w2[(kb + 1) * D_FEAT + n0 + l15];
            acc = __builtin_amdgcn_wmma_f32_16x16x4_f32(
                false, a, false, b, (short)0, acc, false, false);
        }
        const float bias = b2[n0 + l15];
        for (int r = 0; r < 8; ++r) {
            out[(long long)(node0 + r + 8 * hi) * D_FEAT + n0 + l15] = acc[r] + bias;
        }
    }
}

// ---------------------------------------------------------------------------
extern "C" void kernel_launch(void* const* d_in, const int* in_sizes, int n_in,
                              void* d_out, int out_size, void* d_ws, size_t ws_size,
                              hipStream_t stream) {
    const float* x   = (const float*)d_in[0];
    const int*   ei  = (const int*)  d_in[1];
    const float* w1  = (const float*)d_in[2];
    const float* b1  = (const float*)d_in[3];
    const float* w2  = (const float*)d_in[4];
    const float* b2  = (const float*)d_in[5];
    const float* eps = (const float*)d_in[6];
    float* out = (float*)d_out;

    const int n_nodes  = in_sizes[0] / D_FEAT;
    const long long n_edges = in_sizes[1] / 2;
    const int n_feat_elems = n_nodes * D_FEAT;

    float* agg = (float*)d_ws;  // [n_nodes, 64] scratch, rewritten every call

    const int* row = ei;             // edge_index[0]
    const int* col = ei + n_edges;   // edge_index[1]

    // 1) agg = (1+eps) * x
    {
        int blocks = (n_feat_elems + 255) / 256;
        gin_init_agg<<<blocks, 256, 0, stream>>>(x, eps, agg, n_feat_elems);
    }
    // 2) scatter-add neighbor features
    {
        long long n_items = n_edges * D_FEAT;
        long long blocks = (n_items + 255) / 256;
        gin_scatter<<<(int)blocks, 256, 0, stream>>>(x, row, col, agg, n_items);
    }
    // 3) fused WMMA MLP
    {
        int node_tiles = (n_nodes + 15) / 16;
        int blocks = (node_tiles + WAVES_PER_BLOCK - 1) / WAVES_PER_BLOCK;
        gin_mlp<<<blocks, WAVES_PER_BLOCK * 32, 0, stream>>>(
            agg, w1, b1, w2, b2, out, n_nodes);
    }
}